// Occupancy_6571299963201
// MI455X (gfx1250) — compile-verified
//
#include <hip/hip_runtime.h>
#include <hip/hip_bf16.h>

// Problem constants (match reference): C=4, N=131072, S=64, V=131072
#define C_CAM      4
#define N_RAYS     131072
#define S_SEG      64
#define TOTAL_RAYS (C_CAM * N_RAYS)   // 524288
#define RW         16                 // rays processed per wave (pipelined)
#define WPB        8                  // waves per block (256 threads, wave32)
#define NTHREADS   256
#define NBUF       4                  // async ring-buffer depth (3 rays in flight)

// ---------------------------------------------------------------------------
// CDNA5 async global->LDS path (ASYNCcnt). Round-2 asm confirmed
// global_load_async_to_lds_b64 lowers; pointer params are vector-typed.
// ---------------------------------------------------------------------------
#if defined(__HIP_DEVICE_COMPILE__) && \
    __has_builtin(__builtin_amdgcn_global_load_async_to_lds_b64)
#define USE_ASYNC 1
#else
#define USE_ASYNC 0
#endif

#if USE_ASYNC
typedef int v2i __attribute__((vector_size(8)));
typedef __attribute__((address_space(1))) v2i* g_v2i_ptr;
typedef __attribute__((address_space(3))) v2i* l_v2i_ptr;

__device__ __forceinline__ void async_copy_b64(const void* gsrc, void* ldst) {
  __builtin_amdgcn_global_load_async_to_lds_b64(
      (g_v2i_ptr)gsrc, (l_v2i_ptr)ldst, /*imm offset*/ 0, /*cpol*/ 0);
}

#if __has_builtin(__builtin_amdgcn_global_load_async_to_lds_b128)
#define HAVE_B128 1
typedef int v4i __attribute__((vector_size(16)));
typedef __attribute__((address_space(1))) v4i* g_v4i_ptr;
typedef __attribute__((address_space(3))) v4i* l_v4i_ptr;

__device__ __forceinline__ void async_copy_b128(const void* gsrc, void* ldst) {
  __builtin_amdgcn_global_load_async_to_lds_b128(
      (g_v4i_ptr)gsrc, (l_v4i_ptr)ldst, /*imm offset*/ 0, /*cpol*/ 0);
}
#else
#define HAVE_B128 0
#endif

// async instructions issued per staged ray (wave-wide)
constexpr int OPR = HAVE_B128 ? 1 : 2;

template <int N>
__device__ __forceinline__ void wait_asynccnt() {
#if __has_builtin(__builtin_amdgcn_s_wait_asynccnt)
  __builtin_amdgcn_s_wait_asynccnt(N);
#else
  asm volatile("s_wait_asynccnt %0" ::"n"(N) : "memory");
#endif
  asm volatile("" ::: "memory");  // keep LDS reads below the wait
}
#endif

// ---------------------------------------------------------------------------
// Per-ray math: lane l of a wave32 owns segments {2l, 2l+1}.
// weights[s] = alpha[s] * prod_{k=1..s} t[k]   (t shifted: factor at s=0 is 1)
// Inclusive prefix-product scan across lanes (5 shuffle steps), then XOR-tree
// sum. Returns the full per-ray weight sum in every lane.
// ---------------------------------------------------------------------------
__device__ __forceinline__ float ray_accum(int lane,
                                           const float* __restrict__ opacity,
                                           int i0, int i1, float d0, float d1) {
  const float e0 = __expf(-opacity[i0] * d0);   // = 1 - alpha0
  const float e1 = __expf(-opacity[i1] * d1);   // = 1 - alpha1
  const float a0 = 1.0f - e0;
  const float a1 = 1.0f - e1;
  const float t0 = fminf(1.0f, e0 + 1e-10f);    // trans factor, seg 2l
  const float t1 = fminf(1.0f, e1 + 1e-10f);    // trans factor, seg 2l+1

  // Shift-by-one: segment 0's factor is 1 (only lane 0's first element)
  const float x0 = (lane == 0) ? 1.0f : t0;
  const float x1 = t1;

  // Inclusive scan of per-lane pair-products across the wave (wave32)
  float M = x0 * x1;
#pragma unroll
  for (int off = 1; off < 32; off <<= 1) {
    const float v = __shfl_up(M, off, 32);
    if (lane >= off) M *= v;
  }
  float excl = __shfl_up(M, 1, 32);   // exclusive scan = product of lanes < l
  if (lane == 0) excl = 1.0f;

  const float P0 = excl * x0;         // cumprod at segment 2l
  const float P1 = P0 * x1;           // cumprod at segment 2l+1
  float w = a0 * P0 + a1 * P1;

#pragma unroll
  for (int off = 16; off > 0; off >>= 1) w += __shfl_xor(w, off, 32);
  return w;
}

// ---------------------------------------------------------------------------
// Kernel: each wave32 handles RW consecutive rays; leaves/deltas are DMA'd to
// a 4-deep LDS ring buffer (3 rays / 1.5 KB in flight per wave) while the
// current ray is computed.
// ---------------------------------------------------------------------------
__global__ __launch_bounds__(NTHREADS) void occupancy_weights_kernel(
    const float* __restrict__ opacity,   // [V]
    const int*   __restrict__ leaves,    // [C*N*S] int32
    const float* __restrict__ deltas,    // [C*N*S]
    float*       __restrict__ out)       // [C*N]
{
  const int lane = threadIdx.x & 31;
  const int wid  = threadIdx.x >> 5;
  const long long ray0 = ((long long)blockIdx.x * WPB + wid) * RW;

#if USE_ASYNC
  __shared__ __align__(16) int   s_lv[WPB][NBUF][S_SEG];
  __shared__ __align__(16) float s_dl[WPB][NBUF][S_SEG];

  // Stage one ray's 256B of leaves + 256B of deltas into ring slot `buf`.
  auto stage = [&](long long ray, int buf) {
#if HAVE_B128
    // lanes 0..15 copy leaves (16B each), lanes 16..31 copy deltas
    const char* g = (lane < 16)
        ? (const char*)(leaves + (size_t)ray * S_SEG) + 16 * lane
        : (const char*)(deltas + (size_t)ray * S_SEG) + 16 * (lane - 16);
    char* l = (lane < 16)
        ? (char*)(&s_lv[wid][buf][0]) + 16 * lane
        : (char*)(&s_dl[wid][buf][0]) + 16 * (lane - 16);
    async_copy_b128(g, l);
#else
    const size_t base = (size_t)ray * S_SEG + 2 * lane;
    async_copy_b64(leaves + base, &s_lv[wid][buf][2 * lane]);
    async_copy_b64(deltas + base, &s_dl[wid][buf][2 * lane]);
#endif
  };

  // prologue: fill 3 of 4 ring slots
  stage(ray0 + 0, 0);
  stage(ray0 + 1, 1);
  stage(ray0 + 2, 2);

  for (int i = 0; i < RW; ++i) {
    const int buf = i & (NBUF - 1);
    // Issue ray i+3 (overwrites slot of ray i-1, consumed last iteration),
    // then wait until ray i's ops have drained (ASYNCcnt is in-order).
    if (i + 3 < RW) {
      stage(ray0 + i + 3, (i + 3) & (NBUF - 1));
      wait_asynccnt<3 * OPR>();
    } else if (i + 2 < RW) {
      wait_asynccnt<2 * OPR>();
    } else if (i + 1 < RW) {
      wait_asynccnt<1 * OPR>();
    } else {
      wait_asynccnt<0>();
    }

    const int   i0 = s_lv[wid][buf][2 * lane + 0];
    const int   i1 = s_lv[wid][buf][2 * lane + 1];
    const float d0 = s_dl[wid][buf][2 * lane + 0];
    const float d1 = s_dl[wid][buf][2 * lane + 1];

    const float w = ray_accum(lane, opacity, i0, i1, d0, d1);
    if (lane == 0) out[ray0 + i] = w;
  }
#else
  // Fallback: direct coalesced global loads (wave reads 256B/ray contiguous)
  for (int i = 0; i < RW; ++i) {
    const size_t base = (size_t)(ray0 + i) * S_SEG + 2 * lane;
    const int2   lv = *reinterpret_cast<const int2*>(leaves + base);
    const float2 dl = *reinterpret_cast<const float2*>(deltas + base);
    const float w = ray_accum(lane, opacity, lv.x, lv.y, dl.x, dl.y);
    if (lane == 0) out[ray0 + i] = w;
  }
#endif
}

// ---------------------------------------------------------------------------
extern "C" void kernel_launch(void* const* d_in, const int* in_sizes, int n_in,
                              void* d_out, int out_size, void* d_ws,
                              size_t ws_size, hipStream_t stream) {
  const float* opacity = (const float*)d_in[0];  // [V] f32
  const int*   leaves  = (const int*)d_in[1];    // [C,N,S] int32
  const float* deltas  = (const float*)d_in[2];  // [C,N,S] f32
  float*       out     = (float*)d_out;          // [C,N] f32

  // 524288 rays / (8 waves * 16 rays per wave) = 4096 blocks
  const int blocks = TOTAL_RAYS / (WPB * RW);
  occupancy_weights_kernel<<<blocks, NTHREADS, 0, stream>>>(opacity, leaves,
                                                            deltas, out);
  (void)in_sizes; (void)n_in; (void)out_size; (void)d_ws; (void)ws_size;
}